// DAG_RNN_4Neigh_50929722196510
// MI455X (gfx1250) — compile-verified
//
#include <hip/hip_runtime.h>

typedef __attribute__((ext_vector_type(2))) float v2f;
typedef __attribute__((ext_vector_type(8))) float v8f;

#define Bv 32
#define Cv 64
#define Hv 80
#define Wv 80
#define CELL (Cv * Bv)          // 2048 floats = 8 KB per grid cell
#define NTOT (Bv * Cv * Hv * Wv)

// ---- one-time layout transform: x[B][C][H][W] -> xT[H][W][C][B] ----
__global__ void transpose_in_kernel(const float* __restrict__ x,
                                    float* __restrict__ xT) {
  int idx = blockIdx.x * blockDim.x + threadIdx.x;
  if (idx >= NTOT) return;
  int w = idx % Wv;
  int h = (idx / Wv) % Hv;
  int c = (idx / (Wv * Hv)) % Cv;
  int b = idx / (Wv * Hv * Cv);
  xT[(((size_t)h * Wv + w) * Cv + c) * Bv + b] = x[idx];
}

__global__ void zero_kernel(float* __restrict__ p) {
  int idx = blockIdx.x * blockDim.x + threadIdx.x;
  if (idx < NTOT) p[idx] = 0.0f;
}

// ---- final: acc[H][W][C][B] -> out[B][C][H][W] ----
__global__ void final_out_kernel(const float* __restrict__ accT,
                                 float* __restrict__ out) {
  int idx = blockIdx.x * blockDim.x + threadIdx.x;
  if (idx >= NTOT) return;
  int w = idx % Wv;
  int h = (idx / Wv) % Hv;
  int c = (idx / (Wv * Hv)) % Cv;
  int b = idx / (Wv * Hv * Cv);
  out[idx] = accT[(((size_t)h * Wv + w) * Cv + c) * Bv + b];
}

// ---- one anti-diagonal step for all 4 directions ----
// block = 256 threads = 8 waves; one block per (direction, cell); each wave
// owns one 16x16 tile of the [C=64, B=32] cell output and drives
// V_WMMA_F32_16X16X4_F32 over K=64 twice (Gv*up, Gh*left) with two
// independent accumulator chains interleaved for WMMA-level ILP.
__global__ __launch_bounds__(256) void diag_kernel(
    int d, int i_min,
    const float* __restrict__ xT, float* __restrict__ accT,
    float* __restrict__ hring,
    const float* __restrict__ g1, const float* __restrict__ g2,
    const float* __restrict__ g4, const float* __restrict__ g5,
    const float* __restrict__ g7, const float* __restrict__ g8,
    const float* __restrict__ g10, const float* __restrict__ g11) {
  __shared__ float sGv[Cv * Cv];   // 16 KB
  __shared__ float sGh[Cv * Cv];   // 16 KB
  __shared__ float sUp[CELL];      // 8 KB
  __shared__ float sLf[CELL];      // 8 KB

  const int dir = blockIdx.y;          // 0=SE 1=NE 2=NW 3=SW
  const int il = i_min + blockIdx.x;   // logical row on this diagonal
  const int jl = d - il;               // logical col
  const int tid = threadIdx.x;

  const float* Gv; const float* Gh;
  bool flipH, flipW, relu;
  switch (dir) {
    case 0:  Gv = g1;  Gh = g2;  flipH = false; flipW = false; relu = true;  break;
    case 1:  Gv = g4;  Gh = g5;  flipH = true;  flipW = false; relu = true;  break;
    case 2:  Gv = g7;  Gh = g8;  flipH = true;  flipW = true;  relu = true;  break;
    default: Gv = g10; Gh = g11; flipH = false; flipW = true;  relu = false; break;
  }
  const int fi = flipH ? (Hv - 1 - il) : il;   // physical coords
  const int fj = flipW ? (Wv - 1 - jl) : jl;

  // stage gammas into LDS (reused by all 8 waves, 64 k-steps each)
  for (int t = tid; t < Cv * Cv; t += 256) { sGv[t] = Gv[t]; sGh[t] = Gh[t]; }

  // 2-diagonal ring buffer per direction, slot = logical j (L2-resident)
  float* ringDir = hring + (size_t)dir * 2 * Wv * CELL;
  const float* prevRing = ringDir + (size_t)((d ^ 1) & 1) * Wv * CELL;
  float* curRing = ringDir + (size_t)(d & 1) * Wv * CELL;

  if (il > 0) {
    const float* up = prevRing + (size_t)jl * CELL;
    for (int t = tid; t < CELL; t += 256) sUp[t] = up[t];
  } else {
    for (int t = tid; t < CELL; t += 256) sUp[t] = 0.0f;
  }
  if (jl > 0) {
    const float* lf = prevRing + (size_t)(jl - 1) * CELL;
    for (int t = tid; t < CELL; t += 256) sLf[t] = lf[t];
  } else {
    for (int t = tid; t < CELL; t += 256) sLf[t] = 0.0f;
  }
  __syncthreads();

  const int wave = tid >> 5;
  const int lane = tid & 31;
  const int mt = wave >> 1;        // 0..3 : 16-row tile of C dimension
  const int nt = wave & 1;         // 0..1 : 16-col tile of B dimension
  const int hl = lane >> 4;        // lane half (0 | 1)
  const int l15 = lane & 15;

  const size_t cellBase = ((size_t)fi * Wv + fj) * CELL;
  const float* xc = xT + cellBase;

  // C/D layout: VGPR r, lanes0-15 -> M=r, lanes16-31 -> M=r+8
  v8f accV;            // x + Gv*h_up chain
  v8f accH = {};       // Gh*h_left chain (independent for ILP)
  #pragma unroll
  for (int r = 0; r < 8; ++r) {
    int m = mt * 16 + r + hl * 8;
    int n = nt * 16 + l15;
    accV[r] = xc[m * Bv + n];      // h starts as x[cell]
  }

  const int am = mt * 16 + l15;    // A-matrix row (lane<16 and lane>=16 mirror)
  const int bn = nt * 16 + l15;    // B-matrix col

  // interleaved independent WMMA chains: Gv*up -> accV, Gh*left -> accH
  #pragma unroll
  for (int kk = 0; kk < 16; ++kk) {
    int k0 = kk * 4 + hl * 2;      // lanes<16: K0/K1, lanes>=16: K2/K3
    v2f av, bv, ah, bh;
    av.x = sGv[am * Cv + k0];
    av.y = sGv[am * Cv + k0 + 1];
    bv.x = sUp[k0 * Bv + bn];
    bv.y = sUp[(k0 + 1) * Bv + bn];
    ah.x = sGh[am * Cv + k0];
    ah.y = sGh[am * Cv + k0 + 1];
    bh.x = sLf[k0 * Bv + bn];
    bh.y = sLf[(k0 + 1) * Bv + bn];
    accV = __builtin_amdgcn_wmma_f32_16x16x4_f32(false, av, false, bv,
                                                 (short)0, accV, false, false);
    accH = __builtin_amdgcn_wmma_f32_16x16x4_f32(false, ah, false, bh,
                                                 (short)0, accH, false, false);
  }

  #pragma unroll
  for (int r = 0; r < 8; ++r) {
    float v = accV[r] + accH[r];
    accV[r] = relu ? fmaxf(v, 0.0f) : v;
  }

  // publish h for the next diagonal + accumulate into the 4-direction sum
  float* hc = curRing + (size_t)jl * CELL;
  float* oc = accT + cellBase;
  #pragma unroll
  for (int r = 0; r < 8; ++r) {
    int m = mt * 16 + r + hl * 8;
    int n = nt * 16 + l15;
    float v = accV[r];
    hc[m * Bv + n] = v;
    atomicAdd(&oc[m * Bv + n], v);   // SE/NW and NE/SW collide only at d=79
  }
}

extern "C" void kernel_launch(void* const* d_in, const int* in_sizes, int n_in,
                              void* d_out, int out_size, void* d_ws, size_t ws_size,
                              hipStream_t stream) {
  (void)in_sizes; (void)n_in; (void)out_size; (void)ws_size;
  const float* x   = (const float*)d_in[0];
  const float* g1  = (const float*)d_in[1];
  const float* g2  = (const float*)d_in[2];
  const float* g4  = (const float*)d_in[3];
  const float* g5  = (const float*)d_in[4];
  const float* g7  = (const float*)d_in[5];
  const float* g8  = (const float*)d_in[6];
  const float* g10 = (const float*)d_in[7];
  const float* g11 = (const float*)d_in[8];
  float* out = (float*)d_out;

  // workspace: xT (13.1M f32) | accT (13.1M f32) | hring (1.31M f32) ~= 110 MB
  float* xT    = (float*)d_ws;
  float* accT  = xT + (size_t)NTOT;
  float* hring = accT + (size_t)NTOT;

  const int nb = (NTOT + 255) / 256;
  transpose_in_kernel<<<nb, 256, 0, stream>>>(x, xT);
  zero_kernel<<<nb, 256, 0, stream>>>(accT);

  for (int d = 0; d < Hv + Wv - 1; ++d) {
    int imin = d - (Wv - 1); if (imin < 0) imin = 0;
    int imax = d < (Hv - 1) ? d : (Hv - 1);
    int ncells = imax - imin + 1;
    diag_kernel<<<dim3(ncells, 4), 256, 0, stream>>>(
        d, imin, xT, accT, hring, g1, g2, g4, g5, g7, g8, g10, g11);
  }

  final_out_kernel<<<nb, 256, 0, stream>>>(accT, out);
}